// FW_71219147702515
// MI455X (gfx1250) — compile-verified
//
#include <hip/hip_runtime.h>

// Problem constants (match reference)
#define H_  384
#define W_  1280
#define B_  16
#define C_  3
#define HW_ (H_ * W_)          // 491520
#define N_  (B_ * HW_)         // 7864320 source pixels / destination slots
#define CN_ (C_ * N_)          // 23592960 output elements
#define SAME_RANGE 0.2f

// Native Clang 4x32-bit vector: valid for __builtin_nontemporal_* and lowers to b128 VMEM ops.
typedef unsigned int uv4 __attribute__((ext_vector_type(4)));

// ---- order-preserving float<->uint encoding for atomic max over signed floats ----
__device__ __forceinline__ unsigned enc_f32(float f) {
    unsigned u = __float_as_uint(f);
    return (u & 0x80000000u) ? ~u : (u | 0x80000000u);
}
// encode(-inf) = ~0xFF800000 = 0x007FFFFF
#define ENC_NEG_INF 0x007FFFFFu

__device__ __forceinline__ unsigned dec_f32_bits(unsigned e) {
    unsigned u = (e & 0x80000000u) ? (e ^ 0x80000000u) : ~e;  // decode
    if ((u & 0x7F800000u) == 0x7F800000u) u = 0u;             // inf/NaN (holes) -> +0.0f
    return u;
}

__device__ __forceinline__ int dest_index(int b, int hw, float fx, float fy) {
    int y = hw / W_;
    int x = hw - y * W_;
    float sx = fminf(fmaxf((float)x + fx, 0.0f), (float)(W_ - 1));
    float sy = fminf(fmaxf((float)y + fy, 0.0f), (float)(H_ - 1));
    int xi = (int)rintf(sx);           // v_rndne_f32: round-half-to-even == jnp.round
    int yi = (int)rintf(sy);
    return b * HW_ + yi * W_ + xi;     // destination slot within same batch
}

// Pass 0: initialize z-buffer (+inf) and encoded output maxima (encode(-inf)).
// b128 stores: 512 B per wave32 instruction instead of 128 B.
__global__ void fw_init(uv4* __restrict__ dminbits4, uv4* __restrict__ outbits4) {
    int i = blockIdx.x * blockDim.x + threadIdx.x;
    if (i < CN_ / 4) {
        uv4 v = { ENC_NEG_INF, ENC_NEG_INF, ENC_NEG_INF, ENC_NEG_INF };
        outbits4[i] = v;
    }
    if (i < N_ / 4) {
        // +inf bits; depth>=0 so uint order == float order
        uv4 v = { 0x7F800000u, 0x7F800000u, 0x7F800000u, 0x7F800000u };
        dminbits4[i] = v;
    }
}

// Pass 1: z-buffer min via no-return global_atomic_min_u32 on depth bit patterns.
__global__ void fw_zmin(const float* __restrict__ flow,
                        const float* __restrict__ depth,
                        unsigned* __restrict__ dminbits) {
    int s = blockIdx.x * blockDim.x + threadIdx.x;
    if (s >= N_) return;
    int b  = s / HW_;
    int hw = s - b * HW_;
    float fx = flow[(b * 2 + 0) * HW_ + hw];
    float fy = flow[(b * 2 + 1) * HW_ + hw];
    int dst = dest_index(b, hw, fx, fy);
    unsigned dbits = __float_as_uint(depth[s]);   // depth in [0,1): non-negative
    atomicMin(&dminbits[dst], dbits);
}

// Pass 2: depth-gated scatter-max of the 3 channels (encoded uint max -> deterministic).
__global__ void fw_splat(const float* __restrict__ obj,
                         const float* __restrict__ flow,
                         const float* __restrict__ depth,
                         const unsigned* __restrict__ dminbits,
                         unsigned* __restrict__ outbits) {
    int s = blockIdx.x * blockDim.x + threadIdx.x;
    if (s >= N_) return;
    int b  = s / HW_;
    int hw = s - b * HW_;

    // gfx1250 global_prefetch_b8: pull the read-once obj cachelines early.
    __builtin_prefetch(obj + (b * C_) * HW_ + hw, 0, 0);

    float fx = flow[(b * 2 + 0) * HW_ + hw];
    float fy = flow[(b * 2 + 1) * HW_ + hw];
    int dst   = dest_index(b, hw, fx, fy);
    int dsthw = dst - b * HW_;

    float d    = depth[s];
    float dmin = __uint_as_float(dminbits[dst]);  // exact per-destination min
    if (d <= dmin + SAME_RANGE) {
        #pragma unroll
        for (int c = 0; c < C_; ++c) {
            // obj is read exactly once: non-temporal load keeps L2 free for the RMW set.
            float v = __builtin_nontemporal_load(obj + (b * C_ + c) * HW_ + hw);
            atomicMax(&outbits[(b * C_ + c) * HW_ + dsthw], enc_f32(v));
        }
    }
}

// Pass 3: decode encoded maxima in place, 4 at a time; holes -> 0.0f.
// NT on both sides: encoded data is dead after this read, decoded output is
// never re-read by the GPU -> don't occupy L2 lines.
__global__ void fw_finalize(uv4* __restrict__ outbits4) {
    int i = blockIdx.x * blockDim.x + threadIdx.x;
    if (i >= CN_ / 4) return;
    uv4 e = __builtin_nontemporal_load(&outbits4[i]);
    uv4 r;
    r.x = dec_f32_bits(e.x);
    r.y = dec_f32_bits(e.y);
    r.z = dec_f32_bits(e.z);
    r.w = dec_f32_bits(e.w);
    __builtin_nontemporal_store(r, &outbits4[i]);   // bit patterns of float results
}

extern "C" void kernel_launch(void* const* d_in, const int* in_sizes, int n_in,
                              void* d_out, int out_size, void* d_ws, size_t ws_size,
                              hipStream_t stream) {
    const float* obj   = (const float*)d_in[0];   // [B,C,H,W]
    const float* flow  = (const float*)d_in[1];   // [B,2,H,W]
    const float* depth = (const float*)d_in[2];   // [B,1,H,W]

    unsigned* outbits  = (unsigned*)d_out;        // reuse d_out as encoded-max scratch
    unsigned* dminbits = (unsigned*)d_ws;         // N_ * 4 bytes = 31.5 MB of workspace

    const int T = 256;                            // 8 wave32 waves per block
    const int gridN   = (N_ + T - 1) / T;         // 30720 blocks (atomic passes)
    const int gridCN4 = (CN_ / 4 + T - 1) / T;    // 23040 blocks (b128 streaming passes)

    fw_init    <<<gridCN4, T, 0, stream>>>((uv4*)dminbits, (uv4*)outbits);
    fw_zmin    <<<gridN,   T, 0, stream>>>(flow, depth, dminbits);
    fw_splat   <<<gridN,   T, 0, stream>>>(obj, flow, depth, dminbits, outbits);
    fw_finalize<<<gridCN4, T, 0, stream>>>((uv4*)outbits);
}